// HGTGraphUpdate_850403525188
// MI455X (gfx1250) — compile-verified
//
#include <hip/hip_runtime.h>
#include <hip/hip_bf16.h>
#include <math.h>

// ---------------------------------------------------------------------------
// HGT graph-update layer for gfx1250 (MI455X), wave32, WMMA f32 16x16x4.
// Per-head C x C edge projections are folded into node-level GEMM weights,
// so the edge phase is pure gather + dot + softmax + scatter (L2-resident).
// ---------------------------------------------------------------------------

typedef float v2f __attribute__((ext_vector_type(2)));
typedef float v8f __attribute__((ext_vector_type(8)));

#define D_DIM 64
#define H_DIM 8
#define C_DIM 8
#define RSQRT_C 0.35355339059327373f

// -------------------- order-preserving float <-> uint (for atomicMax) ------
__device__ __forceinline__ unsigned enc_f(float f) {
  unsigned u = __float_as_uint(f);
  return (u & 0x80000000u) ? ~u : (u | 0x80000000u);
}
__device__ __forceinline__ float dec_f(unsigned u) {
  return __uint_as_float((u & 0x80000000u) ? (u ^ 0x80000000u) : ~u);
}

// -------------------- K0: fold per-head projections into GEMM weights ------
// Wall layout: [5][64][64] = {Wq, Wk*bd(Watt0), Wm*bd(Wmsg0), Wk*bd(Watt1), Wm*bd(Wmsg1)}
// ball layout: [5][64]     = {bq, bk*bd(Watt0), bm*bd(Wmsg0), bk*bd(Watt1), bm*bd(Wmsg1)}
__global__ void build_weights_kernel(
    const float* __restrict__ Wk, const float* __restrict__ bk,
    const float* __restrict__ Wm, const float* __restrict__ bm,
    const float* __restrict__ Wq, const float* __restrict__ bq,
    const float* __restrict__ Watt0, const float* __restrict__ Wmsg0,
    const float* __restrict__ Watt1, const float* __restrict__ Wmsg1,
    float* __restrict__ Wall, float* __restrict__ ball) {
  const int t = threadIdx.x;  // 256 threads, 1 block
  for (int i = t; i < 4096; i += 256) Wall[i] = Wq[i];
  if (t < 64) ball[t] = bq[t];

  const float* Ws[4] = {Wk, Wm, Wk, Wm};
  const float* bs[4] = {bk, bm, bk, bm};
  const float* Hs[4] = {Watt0, Wmsg0, Watt1, Wmsg1};
  for (int mm = 0; mm < 4; ++mm) {
    const float* W = Ws[mm];
    const float* Hh = Hs[mm];
    for (int i = t; i < 4096; i += 256) {
      int d = i >> 6, o = i & 63;
      int h = o >> 3, l = o & 7;
      float s = 0.f;
#pragma unroll
      for (int kk = 0; kk < 8; ++kk)
        s += W[d * 64 + h * 8 + kk] * Hh[h * 64 + kk * 8 + l];
      Wall[(mm + 1) * 4096 + i] = s;
    }
    if (t < 64) {
      int h = t >> 3, l = t & 7;
      float s = 0.f;
      const float* b = bs[mm];
#pragma unroll
      for (int kk = 0; kk < 8; ++kk)
        s += b[h * 8 + kk] * Hh[h * 64 + kk * 8 + l];
      ball[(mm + 1) * 64 + t] = s;
    }
  }
}

// -------------------- K1/K5b: node GEMM via V_WMMA_F32_16X16X4_F32 ---------
// out[mat] = X @ Wall[mat] + ball[mat], X:[N,64], W:[64,64].
// Block = 256 thr = 8 waves; wave owns a 16-row tile; A-frags preloaded in
// regs, B-frags from LDS; 16 chained WMMAs per 16x16 output tile.
__global__ void gemm_nodes_wmma_kernel(
    const float* __restrict__ X, const float* __restrict__ Wall,
    const float* __restrict__ ball,
    float* __restrict__ out0, float* __restrict__ out1, float* __restrict__ out2,
    float* __restrict__ out3, float* __restrict__ out4, int nm, int N) {
  __shared__ float Wlds[64 * 64];
  __shared__ float blds[64];
  float* outs[5] = {out0, out1, out2, out3, out4};

  const int lane = threadIdx.x & 31;
  const int wave = threadIdx.x >> 5;
  const int r0 = (blockIdx.x * 8 + wave) * 16;
  const int mrow = lane & 15;            // A-matrix row / B,D column-in-tile
  const int koff = (lane >> 4) * 2;      // K sub-offset per half-wave
  const int row = r0 + mrow;
  const bool rv = row < N;

  // Preload full 16x64 A tile as 16 fragments (v2f each).
  v2f afrag[16];
#pragma unroll
  for (int ks = 0; ks < 16; ++ks) {
    afrag[ks].x = rv ? X[(size_t)row * 64 + ks * 4 + koff + 0] : 0.f;
    afrag[ks].y = rv ? X[(size_t)row * 64 + ks * 4 + koff + 1] : 0.f;
  }

  for (int mat = 0; mat < nm; ++mat) {
    __syncthreads();
    for (int i = threadIdx.x; i < 64 * 64; i += blockDim.x)
      Wlds[i] = Wall[mat * 4096 + i];
    if (threadIdx.x < 64) blds[threadIdx.x] = ball[mat * 64 + threadIdx.x];
    __syncthreads();

    float* outp = outs[mat];
    for (int ct = 0; ct < 4; ++ct) {
      v8f c = {};
#pragma unroll
      for (int ks = 0; ks < 16; ++ks) {
        v2f b;
        b.x = Wlds[(ks * 4 + koff + 0) * 64 + ct * 16 + mrow];
        b.y = Wlds[(ks * 4 + koff + 1) * 64 + ct * 16 + mrow];
        c = __builtin_amdgcn_wmma_f32_16x16x4_f32(
            /*neg_a=*/false, afrag[ks], /*neg_b=*/false, b,
            /*c_mod=*/(short)0, c, /*reuse_a=*/false, /*reuse_b=*/false);
      }
      const int col = ct * 16 + mrow;
      const float bias = blds[col];
      const int rbase = r0 + ((lane >> 4) << 3);  // lanes 16..31 -> rows +8
#pragma unroll
      for (int i = 0; i < 8; ++i) {
        int rr = rbase + i;
        if (rr < N) outp[(size_t)rr * 64 + col] = c[i] + bias;
      }
    }
  }
}

// -------------------- K2: per-(edge,head) attention scores + segment max ---
__global__ void edge_scores_kernel(
    const int* __restrict__ src, const int* __restrict__ dst,
    const float* __restrict__ ka, const float* __restrict__ q,
    const float* __restrict__ prior, float* __restrict__ escore,
    unsigned* __restrict__ nmax, int E, int ebase) {
  int t = blockIdx.x * blockDim.x + threadIdx.x;
  int e = t >> 3;
  int h = t & 7;
  if (e >= E) return;
  int s = src[e], d = dst[e];
  const float4* ap = (const float4*)(ka + (size_t)s * 64 + h * 8);
  const float4* qp = (const float4*)(q + (size_t)d * 64 + h * 8);
  float4 a0 = ap[0], a1 = ap[1], q0 = qp[0], q1 = qp[1];
  float dot = a0.x * q0.x + a0.y * q0.y + a0.z * q0.z + a0.w * q0.w +
              a1.x * q1.x + a1.y * q1.y + a1.z * q1.z + a1.w * q1.w;
  float sc = dot * prior[h] * RSQRT_C;
  escore[(size_t)(ebase + e) * 8 + h] = sc;
  atomicMax(nmax + (size_t)d * 8 + h, enc_f(sc));
}

// -------------------- K3: exp(score - max) + segment denom -----------------
__global__ void edge_exp_kernel(
    const int* __restrict__ dst0, const int* __restrict__ dst1,
    float* __restrict__ escore, const unsigned* __restrict__ nmax,
    float* __restrict__ denom, int E) {
  int t = blockIdx.x * blockDim.x + threadIdx.x;
  if (t >= 2 * E * 8) return;
  int e = t >> 3, h = t & 7;
  int d = (e < E) ? dst0[e] : dst1[e - E];
  float mx = dec_f(nmax[(size_t)d * 8 + h]);
  float ex = expf(escore[t] - mx);
  escore[t] = ex;
  atomicAdd(denom + (size_t)d * 8 + h, ex);
}

// -------------------- K4: weighted message scatter (wave per edge) ---------
__global__ void edge_scatter_kernel(
    const int* __restrict__ src0, const int* __restrict__ dst0,
    const int* __restrict__ src1, const int* __restrict__ dst1,
    const float* __restrict__ ma0, const float* __restrict__ ma1,
    const float* __restrict__ escore, const float* __restrict__ denom,
    float* __restrict__ pooled, int E) {
  int wid = (int)((blockIdx.x * blockDim.x + threadIdx.x) >> 5);
  int lane = threadIdx.x & 31;
  if (wid >= 2 * E) return;
  int s, d;
  const float* ma;
  if (wid < E) {
    s = src0[wid]; d = dst0[wid]; ma = ma0;
  } else {
    int e2 = wid - E;
    s = src1[e2]; d = dst1[e2]; ma = ma1;
  }
  int h = lane >> 2;  // channels 2*lane, 2*lane+1 -> head (2*lane)/8
  float w = escore[(size_t)wid * 8 + h] / denom[(size_t)d * 8 + h];
  float2 mv = ((const float2*)(ma + (size_t)s * 64))[lane];
  atomicAdd(pooled + (size_t)d * 64 + lane * 2 + 0, w * mv.x);
  atomicAdd(pooled + (size_t)d * 64 + lane * 2 + 1, w * mv.y);
}

// -------------------- K5a: exact gelu --------------------------------------
__global__ void gelu_kernel(const float* __restrict__ in, float* __restrict__ out,
                            int n) {
  int t = blockIdx.x * blockDim.x + threadIdx.x;
  if (t >= n) return;
  float v = in[t];
  out[t] = 0.5f * v * (1.f + erff(v * 0.70710678118654752f));
}

// -------------------- K5c: weighted skip + LayerNorm (wave per node) -------
__global__ void skip_ln_kernel(
    const float* __restrict__ agg, const float* __restrict__ x,
    const float* __restrict__ skip_w, const float* __restrict__ gamma,
    const float* __restrict__ beta, float* __restrict__ out, int N) {
  int wid = (int)((blockIdx.x * blockDim.x + threadIdx.x) >> 5);
  int lane = threadIdx.x & 31;
  if (wid >= N) return;
  float sc = 1.f / (1.f + expf(-skip_w[0]));
  float2 av = ((const float2*)(agg + (size_t)wid * 64))[lane];
  float2 xv = ((const float2*)(x + (size_t)wid * 64))[lane];
  float2 v;
  v.x = sc * av.x + (1.f - sc) * xv.x;
  v.y = sc * av.y + (1.f - sc) * xv.y;
  float s1 = v.x + v.y;
  float s2 = v.x * v.x + v.y * v.y;
#pragma unroll
  for (int off = 16; off >= 1; off >>= 1) {
    s1 += __shfl_xor(s1, off, 32);
    s2 += __shfl_xor(s2, off, 32);
  }
  float mean = s1 * (1.f / 64.f);
  float var = s2 * (1.f / 64.f) - mean * mean;
  float r = rsqrtf(var + 1e-3f);
  int c = lane * 2;
  out[(size_t)wid * 64 + c + 0] = gamma[c + 0] * (v.x - mean) * r + beta[c + 0];
  out[(size_t)wid * 64 + c + 1] = gamma[c + 1] * (v.y - mean) * r + beta[c + 1];
}

// ---------------------------------------------------------------------------
extern "C" void kernel_launch(void* const* d_in, const int* in_sizes, int n_in,
                              void* d_out, int out_size, void* d_ws,
                              size_t ws_size, hipStream_t stream) {
  const float* x     = (const float*)d_in[0];
  const int* src0    = (const int*)d_in[1];
  const int* dst0    = (const int*)d_in[2];
  const int* src1    = (const int*)d_in[3];
  const int* dst1    = (const int*)d_in[4];
  const float* Wk    = (const float*)d_in[5];
  const float* bk    = (const float*)d_in[6];
  const float* Wm    = (const float*)d_in[7];
  const float* bm    = (const float*)d_in[8];
  const float* Wq    = (const float*)d_in[9];
  const float* bq    = (const float*)d_in[10];
  const float* Wa    = (const float*)d_in[11];
  const float* ba    = (const float*)d_in[12];
  const float* Watt0 = (const float*)d_in[13];
  const float* Wmsg0 = (const float*)d_in[14];
  const float* prior0= (const float*)d_in[15];
  const float* Watt1 = (const float*)d_in[16];
  const float* Wmsg1 = (const float*)d_in[17];
  const float* prior1= (const float*)d_in[18];
  const float* skipw = (const float*)d_in[19];
  const float* gamma = (const float*)d_in[20];
  const float* beta  = (const float*)d_in[21];

  const int N = in_sizes[0] / D_DIM;
  const int E = in_sizes[1];
  const size_t N64 = (size_t)N * 64;

  // ---- workspace carve-up (floats) ----
  float* w = (float*)d_ws;
  float* qv     = w;            w += N64;
  float* ka0    = w;            w += N64;
  float* ma0    = w;            w += N64;
  float* ka1    = w;            w += N64;
  float* ma1    = w;            w += N64;
  float* pooled = w;            w += N64;       // later reused as `agg`
  float* Wall   = w;            w += 5 * 4096;
  float* ball   = w;            w += 5 * 64;
  float* denom  = w;            w += (size_t)N * 8;
  unsigned* nmax = (unsigned*)w; w += (size_t)N * 8;
  float* escore = w;            w += (size_t)2 * E * 8;
  // gelu buffer: reuse escore region if it is large enough, else fresh
  float* gbuf = ((size_t)2 * E * 8 >= N64) ? escore : w;
  float* agg  = pooled;  // pooled dead after gelu

  // ---- zero accumulators (graph-capture safe) ----
  hipMemsetAsync(pooled, 0, N64 * sizeof(float), stream);
  hipMemsetAsync(denom, 0, (size_t)N * 8 * sizeof(float), stream);
  hipMemsetAsync(nmax, 0, (size_t)N * 8 * sizeof(unsigned), stream);  // enc(-inf)

  // ---- K0: effective weights ----
  build_weights_kernel<<<1, 256, 0, stream>>>(Wk, bk, Wm, bm, Wq, bq, Watt0,
                                              Wmsg0, Watt1, Wmsg1, Wall, ball);

  // ---- K1: 5 fused node GEMMs (WMMA) ----
  int gemm_blocks = (N + 127) / 128;
  gemm_nodes_wmma_kernel<<<gemm_blocks, 256, 0, stream>>>(
      x, Wall, ball, qv, ka0, ma0, ka1, ma1, 5, N);

  // ---- K2: scores + segment max ----
  int sc_blocks = (E * 8 + 255) / 256;
  edge_scores_kernel<<<sc_blocks, 256, 0, stream>>>(src0, dst0, ka0, qv, prior0,
                                                    escore, nmax, E, 0);
  edge_scores_kernel<<<sc_blocks, 256, 0, stream>>>(src1, dst1, ka1, qv, prior1,
                                                    escore, nmax, E, E);

  // ---- K3: exp + denom ----
  int ex_blocks = (2 * E * 8 + 255) / 256;
  edge_exp_kernel<<<ex_blocks, 256, 0, stream>>>(dst0, dst1, escore, nmax,
                                                 denom, E);

  // ---- K4: weighted message scatter ----
  int scat_blocks = (2 * E + 7) / 8;  // wave per edge, 8 waves/block
  edge_scatter_kernel<<<scat_blocks, 256, 0, stream>>>(
      src0, dst0, src1, dst1, ma0, ma1, escore, denom, pooled, E);

  // ---- K5a: gelu ----
  int g_blocks = (int)((N64 + 255) / 256);
  gelu_kernel<<<g_blocks, 256, 0, stream>>>(pooled, gbuf, (int)N64);

  // ---- K5b: agg = gelu(pooled) @ Wa + ba (WMMA) ----
  gemm_nodes_wmma_kernel<<<gemm_blocks, 256, 0, stream>>>(
      gbuf, Wa, ba, agg, agg, agg, agg, agg, 1, N);

  // ---- K5c: weighted skip + LayerNorm ----
  int ln_blocks = (N * 32 + 255) / 256;
  skip_ln_kernel<<<ln_blocks, 256, 0, stream>>>(agg, x, skipw, gamma, beta,
                                                (float*)d_out, N);
}